// Quantization_39273180954636
// MI455X (gfx1250) — compile-verified
//
#include <hip/hip_runtime.h>
#include <hip/hip_bf16.h>

typedef float v2f __attribute__((ext_vector_type(2)));
typedef float v8f __attribute__((ext_vector_type(8)));

constexpr int D_   = 768;   // embedding size
constexpr int P_   = 96;    // partitions
constexpr int C_   = 256;   // centroids per partition
constexpr int SD_  = 8;     // sub-vector dim
constexpr int ROWS = 128;   // batch rows per block

// One block = one partition p x 128 batch rows. 256 threads = 8 wave32.
__global__ __launch_bounds__(256, 1)
void pq_fwd_wmma(const float* __restrict__ vecs,
                 const float* __restrict__ codebook,
                 float* __restrict__ out)
{
    __shared__ float s_cb[C_ * SD_];    // 8 KB: codebook[p] row-major [c][k]
    __shared__ float s_hcs[C_];         // 1 KB: 0.5*||c||^2
    __shared__ float s_v[ROWS * SD_];   // 4 KB: vecs tile [row][k]
    __shared__ int   s_idx[ROWS];       // winning centroid per row

    const int tid = threadIdx.x;
    const int p   = blockIdx.y;
    const int b0  = blockIdx.x * ROWS;

    // ---- stage codebook[p]: 2048 floats, 2 float4 per thread (coalesced) ----
    {
        const float4* src = (const float4*)(codebook + (size_t)p * C_ * SD_);
        float4* dst = (float4*)s_cb;
        dst[tid]       = src[tid];
        dst[tid + 256] = src[tid + 256];
    }
    // ---- stage vecs slice [b0..b0+127][p*8..p*8+7]: 256 float4 ----
    {
        const int row = tid >> 1, half = tid & 1;
        ((float4*)s_v)[tid] =
            *(const float4*)(vecs + (size_t)(b0 + row) * D_ + p * SD_ + half * 4);
    }
    __syncthreads();

    // ---- 0.5*||c||^2 per centroid (one thread per centroid) ----
    {
        const float* c = s_cb + tid * SD_;
        float acc = 0.f;
        #pragma unroll
        for (int k = 0; k < SD_; ++k) acc += c[k] * c[k];
        s_hcs[tid] = 0.5f * acc;
    }
    __syncthreads();

    const int lane = tid & 31;
    const int wave = tid >> 5;
    const int lh   = lane >> 4;     // lane half: 0 (lanes 0-15) / 1 (lanes 16-31)
    const int ln   = lane & 15;
    const int m0   = wave * 16;     // this wave's row-tile base

    // A operand, 16x4 f32 layout: lane holds M = ln, VGPR j holds K = j + 2*lh.
    // SD=8 split into two K=4 chunks (a0: K 0..3, a1: K 4..7).
    const float* vr = s_v + (m0 + ln) * SD_;
    const v2f a0 = *(const v2f*)(vr + 2 * lh);
    const v2f a1 = *(const v2f*)(vr + 4 + 2 * lh);

    float best[8];
    int   bidx[8];
    #pragma unroll
    for (int k = 0; k < 8; ++k) { best[k] = -3.402823e38f; bidx[k] = 0; }

    // ---- sweep 16 column tiles of centroids: score = v.c - 0.5||c||^2 ----
    #pragma unroll
    for (int t = 0; t < 16; ++t) {
        const int n = t * 16 + ln;                   // this lane's centroid column
        const float* cbn = s_cb + n * SD_;
        // B operand mirrors A's slot->K mapping (4x16: lane holds N=ln, K=j+2*lh)
        const v2f bA = *(const v2f*)(cbn + 2 * lh);
        const v2f bB = *(const v2f*)(cbn + 4 + 2 * lh);
        const float ci = -s_hcs[n];                  // seed accumulator with -0.5||c||^2
        v8f acc = { ci, ci, ci, ci, ci, ci, ci, ci };
        acc = __builtin_amdgcn_wmma_f32_16x16x4_f32(
                  false, a0, false, bA, (short)0, acc, false, false);
        acc = __builtin_amdgcn_wmma_f32_16x16x4_f32(
                  false, a1, false, bB, (short)0, acc, false, false);
        #pragma unroll
        for (int k = 0; k < 8; ++k) {                // C/D: VGPR k -> row m0 + k + 8*lh
            const float s = acc[k];
            if (s > best[k]) { best[k] = s; bidx[k] = n; }
        }
    }

    // ---- argmax across the 16 lanes of each half (columns n = t*16 + ln) ----
    #pragma unroll
    for (int off = 1; off < 16; off <<= 1) {
        #pragma unroll
        for (int k = 0; k < 8; ++k) {
            const float os = __shfl_xor(best[k], off, 32);
            const int   oi = __shfl_xor(bidx[k], off, 32);
            if (os > best[k] || (os == best[k] && oi < bidx[k])) {
                best[k] = os; bidx[k] = oi;
            }
        }
    }

    if (ln == 0) {                                   // lanes 0 and 16 hold the results
        #pragma unroll
        for (int k = 0; k < 8; ++k)
            s_idx[m0 + 8 * lh + k] = bidx[k];
    }
    __syncthreads();

    // ---- gather winning centroids to output: 2 threads x float4 per row ----
    {
        const int row = tid >> 1, half = tid & 1;
        const int idx = s_idx[row];
        const float4 val = *(const float4*)(s_cb + idx * SD_ + half * 4);
        *(float4*)(out + (size_t)(b0 + row) * D_ + p * SD_ + half * 4) = val;
    }
}

extern "C" void kernel_launch(void* const* d_in, const int* in_sizes, int n_in,
                              void* d_out, int out_size, void* d_ws, size_t ws_size,
                              hipStream_t stream)
{
    const float* vecs     = (const float*)d_in[0];   // [4096, 768] f32
    const float* codebook = (const float*)d_in[1];   // [96, 256, 8] f32
    float* out            = (float*)d_out;           // [4096, 768] f32

    dim3 grid(4096 / ROWS, P_);   // 32 x 96 blocks
    dim3 block(256);
    pq_fwd_wmma<<<grid, block, 0, stream>>>(vecs, codebook, out);
}